// PGDCorrection_28750511079900
// MI455X (gfx1250) — compile-verified
//
#include <hip/hip_runtime.h>
#include <hip/hip_bf16.h>
#include <math.h>

// ---------------------------------------------------------------------------
// PGD correction pipeline for MI455X (gfx1250, wave32).
//   mesh 128^3 f32, NPART = 2*128^3 particles.
//   FFTs = batched complex DFT matmuls on V_WMMA_F32_16X16X4_F32 (the only
//   full-precision f32 WMMA shape). Mesh is L2-resident (8-16MB vs 192MB L2),
//   so DFT passes are matrix-pipe bound; particle I/O (~150MB) ~6us at
//   23.3 TB/s. Twiddle B-slab staged to LDS once per workgroup via the
//   Tensor Data Mover and shared by 8 waves; strides are template constants
//   so the fully-unrolled K-loop uses immediate-offset global/ds loads with
//   no per-iteration address arithmetic; each wave carries two 16x16 complex
//   accumulator tiles so every B-fragment feeds 8 WMMAs.
// ---------------------------------------------------------------------------

#define MESH_N   128
#define MESH_N2  (128 * 128)          // 16384
#define MESH_N3  (128 * 128 * 128)    // 2097152
#define NPART    (2 * MESH_N3)        // 4194304
#define TWO_PI_F 6.283185307179586f

typedef float v2f  __attribute__((ext_vector_type(2)));
typedef float v8f  __attribute__((ext_vector_type(8)));
typedef unsigned int u32x4 __attribute__((ext_vector_type(4)));
typedef int   i32x4 __attribute__((ext_vector_type(4)));
typedef int   i32x8 __attribute__((ext_vector_type(8)));

#if defined(__gfx1250__) && __has_builtin(__builtin_amdgcn_tensor_load_to_lds)
#define PGD_USE_TDM 1
#else
#define PGD_USE_TDM 0
#endif

// ---------------------------------------------------------------------------
// Twiddle tables: Wf[j*128+k] = exp(-2*pi*i*j*k/128)
//                 Wi[j*128+k] = exp(+2*pi*i*j*k/128) / 128   (per-axis 1/N)
// ---------------------------------------------------------------------------
__global__ __launch_bounds__(256) void twiddle_kernel(float* __restrict__ wfr,
                                                      float* __restrict__ wfi,
                                                      float* __restrict__ wir,
                                                      float* __restrict__ wii) {
    int idx = blockIdx.x * blockDim.x + threadIdx.x;
    if (idx >= MESH_N2) return;
    int j = idx >> 7;
    int k = idx & 127;
    int t = (j * k) & 127;                       // exp is periodic mod 128
    float a = -TWO_PI_F * (float)t * (1.0f / 128.0f);
    float c = cosf(a), s = sinf(a);
    wfr[idx] = c;
    wfi[idx] = s;
    wir[idx] = c * (1.0f / 128.0f);              // cos(-a)=cos(a)
    wii[idx] = -s * (1.0f / 128.0f);             // sin(-a)=-sin(a)
}

__global__ __launch_bounds__(256) void zero_kernel(float* __restrict__ p, int n) {
    int idx = blockIdx.x * blockDim.x + threadIdx.x;
    if (idx < n) p[idx] = 0.0f;
}

// ---------------------------------------------------------------------------
// CIC paint: 8-corner scatter-add (global_atomic_add_f32).
// _CONN order = itertools.product((0,1),repeat=3): bit2->x, bit1->y, bit0->z.
// ---------------------------------------------------------------------------
__global__ __launch_bounds__(256) void cic_paint_kernel(const float* __restrict__ pos,
                                                        float* __restrict__ mesh) {
    int p = blockIdx.x * blockDim.x + threadIdx.x;
    if (p >= NPART) return;
    float px = pos[3 * p + 0];
    float py = pos[3 * p + 1];
    float pz = pos[3 * p + 2];
    float fx = floorf(px), fy = floorf(py), fz = floorf(pz);
#pragma unroll
    for (int c = 0; c < 8; ++c) {
        float nx = fx + (float)((c >> 2) & 1);
        float ny = fy + (float)((c >> 1) & 1);
        float nz = fz + (float)(c & 1);
        float w = (1.0f - fabsf(px - nx)) * (1.0f - fabsf(py - ny)) * (1.0f - fabsf(pz - nz));
        int ix = ((int)nx) & 127;
        int iy = ((int)ny) & 127;
        int iz = ((int)nz) & 127;
        atomicAdd(&mesh[(ix << 14) + (iy << 7) + iz], w);
    }
}

// ---------------------------------------------------------------------------
// TDM descriptor helper: 2-D tile load, 4-byte elements, tile 16 x 128 from a
// 128x128 f32 plane (row stride 128), into contiguous LDS [k][n] (16 floats
// per row). See CDNA5 ISA (D# groups).
// ---------------------------------------------------------------------------
#if PGD_USE_TDM
__device__ __forceinline__ void tdm_load_w_slab(const float* gsrc, void* lds_dst) {
    unsigned long long ga = (unsigned long long)(uintptr_t)gsrc;
    u32x4 g0;
    g0[0] = 1u;                                         // count=1, user mode
    g0[1] = (unsigned)(uintptr_t)lds_dst;               // lds_addr (bytes)
    g0[2] = (unsigned)(ga & 0xFFFFFFFFu);               // global_addr[31:0]
    g0[3] = (unsigned)((ga >> 32) & 0x01FFFFFFu) | (2u << 30);  // addr[56:32] | type=2
    i32x8 g1;
    g1[0] = 0x00020000;          // workgroup_mask=0, data_size=2 (4B), no flags
    g1[1] = (MESH_N & 0xFFFF) << 16;   // atomic_barrier_addr=0 | tensor_dim0.lo16
    g1[2] = (MESH_N & 0xFFFF) << 16;   // tensor_dim0.hi16=0    | tensor_dim1.lo16
    g1[3] = 16 << 16;                  // tensor_dim1.hi16=0    | tile_dim0=16
    g1[4] = MESH_N;                    // tile_dim1=128         | tile_dim2=0
    g1[5] = MESH_N;                    // tensor_dim0_stride[31:0]=128
    g1[6] = 0;                         // stride0.hi16 | tensor_dim1_stride.lo16
    g1[7] = 0;
    i32x4 gz = {0, 0, 0, 0};           // 2-D tensor: groups 2/3 unused
#if __clang_major__ >= 23
    i32x8 gz8 = {0, 0, 0, 0, 0, 0, 0, 0};
    __builtin_amdgcn_tensor_load_to_lds(g0, g1, gz, gz, gz8, 0);
#else
    __builtin_amdgcn_tensor_load_to_lds(g0, g1, gz, gz, 0);
#endif
}
#endif

// ---------------------------------------------------------------------------
// Batched complex DFT along one axis via V_WMMA_F32_16X16X4_F32.
//   rows r in [0,16384): base(r) = (r>>7)*OUTS + (r&127)*INNS
//   out[base + n*AXS] = sum_k in[base + k*AXS] * W[k*128 + n]
// Block = 256 threads = 8 waves: one column-tile (16 outputs), 16 row-tiles
// (2 per wave: t = rowgrp*16 + wave and t+8, whose bases differ by exactly
// OUTS -> constant immediate offsets). Twiddle slab (128K x 16N, re+im =
// 16KB) staged in LDS (TDM when available), reused by all 8 waves.
// Complex tile math: Cre += Are*Bre + (-Aim)*Bim ; Cim += Are*Bim + Aim*Bre
// (A-negate in VALU: float WMMA NEG bits only support C-negate.)
// A layout (f32 16x4): lane m = l&15, k = 2*(l>>4)+j. B mirrors with n = l&15.
// C layout (f32 16x16): VGPR v: m = v + 8*(l>>4), n = l&15.
// ---------------------------------------------------------------------------
template <int AXS, int OUTS, int INNS, bool HASIM, bool STOREIM>
__global__ __launch_bounds__(256) void dft_pass(const float* __restrict__ inRe,
                                                const float* __restrict__ inIm,
                                                float* __restrict__ outRe,
                                                float* __restrict__ outIm,
                                                const float* __restrict__ Wre,
                                                const float* __restrict__ Wim) {
    __shared__ float sRe[MESH_N * 16];   // [k][n], 8KB
    __shared__ float sIm[MESH_N * 16];   // [k][n], 8KB

    const int tid      = threadIdx.x;
    const int col_tile = blockIdx.x & 7;      // 128/16 = 8 column tiles
    const int rowgrp   = blockIdx.x >> 3;     // 64 groups of 16 row-tiles
    const int cb       = col_tile * 16;

#if PGD_USE_TDM
    if (tid < 32) {                            // wave 0 drives the TDM
        tdm_load_w_slab(Wre + cb, sRe);
        tdm_load_w_slab(Wim + cb, sIm);
        __builtin_amdgcn_s_wait_tensorcnt(0);
    }
#else
#pragma unroll
    for (int i = 0; i < 8; ++i) {
        int e = tid + i * 256;                 // e = k*16 + n
        sRe[e] = Wre[(e >> 4) * MESH_N + cb + (e & 15)];
        sIm[e] = Wim[(e >> 4) * MESH_N + cb + (e & 15)];
    }
#endif
    __syncthreads();

    const int lane = tid & 31;
    const int wave = tid >> 5;
    const int hi = lane >> 4;
    const int lo = lane & 15;

    // Tile 0: row_tile t0 = rowgrp*16 + wave (wave<8 -> t0>>3 = 2*rowgrp,
    // t0&7 = wave). Tile 1: t0+8 -> base differs by exactly +OUTS.
    const int baseA = 2 * rowgrp * OUTS + (wave * 16 + lo) * INNS;

    const float* pRe = inRe + baseA + 2 * hi * AXS;
    const float* pIm = (HASIM ? inIm : inRe) + baseA + 2 * hi * AXS;
    const int bidx = 2 * hi * 16 + lo;        // LDS index of B.x at kb=0

    v8f c0re = {}, c0im = {}, c1re = {}, c1im = {};

#pragma unroll
    for (int kb = 0; kb < 128; kb += 4) {
        v2f a0, a1, bRe, bIm;
        a0.x = pRe[kb * AXS];
        a0.y = pRe[kb * AXS + AXS];
        a1.x = pRe[kb * AXS + OUTS];
        a1.y = pRe[kb * AXS + AXS + OUTS];
        bRe.x = sRe[bidx + kb * 16];
        bRe.y = sRe[bidx + kb * 16 + 16];
        bIm.x = sIm[bidx + kb * 16];
        bIm.y = sIm[bidx + kb * 16 + 16];

        c0re = __builtin_amdgcn_wmma_f32_16x16x4_f32(false, a0, false, bRe,
                                                     (short)0, c0re, false, false);
        c0im = __builtin_amdgcn_wmma_f32_16x16x4_f32(false, a0, false, bIm,
                                                     (short)0, c0im, false, false);
        c1re = __builtin_amdgcn_wmma_f32_16x16x4_f32(false, a1, false, bRe,
                                                     (short)0, c1re, false, false);
        c1im = __builtin_amdgcn_wmma_f32_16x16x4_f32(false, a1, false, bIm,
                                                     (short)0, c1im, false, false);
        if (HASIM) {                          // compile-time: no runtime branch
            v2f i0, i1, n0, n1;
            i0.x = pIm[kb * AXS];
            i0.y = pIm[kb * AXS + AXS];
            i1.x = pIm[kb * AXS + OUTS];
            i1.y = pIm[kb * AXS + AXS + OUTS];
            n0.x = -i0.x; n0.y = -i0.y;
            n1.x = -i1.x; n1.y = -i1.y;
            c0re = __builtin_amdgcn_wmma_f32_16x16x4_f32(false, n0, false, bIm,
                                                         (short)0, c0re, false, false);
            c0im = __builtin_amdgcn_wmma_f32_16x16x4_f32(false, i0, false, bRe,
                                                         (short)0, c0im, false, false);
            c1re = __builtin_amdgcn_wmma_f32_16x16x4_f32(false, n1, false, bIm,
                                                         (short)0, c1re, false, false);
            c1im = __builtin_amdgcn_wmma_f32_16x16x4_f32(false, i1, false, bRe,
                                                         (short)0, c1im, false, false);
        }
    }

    // Stores: tile-0 row rr = rowgrp*16+wave tile -> ob; tile-1 at ob+OUTS.
    const int nCol = cb + lo;
    float* oRe = outRe + nCol * AXS;
    float* oIm = outIm + nCol * AXS;
#pragma unroll
    for (int v = 0; v < 8; ++v) {
        const int m  = v + 8 * hi;
        const int ob = 2 * rowgrp * OUTS + (wave * 16 + m) * INNS;
        oRe[ob]        = c0re[v];
        oRe[ob + OUTS] = c1re[v];
        if (STOREIM) {
            oIm[ob]        = c0im[v];
            oIm[ob + OUTS] = c1im[v];
        }
    }
}

// ---------------------------------------------------------------------------
// Spectral kernels. kvec[d] = 2*pi*fftfreq(128) along axis d.
// ---------------------------------------------------------------------------
__device__ __forceinline__ float kfreq(int j) {
    int s = (j < 64) ? j : j - 128;
    return (float)s * (TWO_PI_F / 128.0f);
}

// pot_k = delta_k * invlaplace(kk) * pgd(kk, kl, ks), in place.
__global__ __launch_bounds__(256) void pot_kernel(float* __restrict__ re,
                                                  float* __restrict__ im,
                                                  const float* __restrict__ klp,
                                                  const float* __restrict__ ksp) {
    int idx = blockIdx.x * blockDim.x + threadIdx.x;
    if (idx >= MESH_N3) return;
    int z = idx & 127, y = (idx >> 7) & 127, x = idx >> 14;
    float kx = kfreq(x), ky = kfreq(y), kz = kfreq(z);
    float kk = kx * kx + ky * ky + kz * kz;
    float s = 0.0f;
    if (kk != 0.0f) {
        float kl = klp[0], ks = ksp[0];
        float ks4 = (ks * ks) * (ks * ks);
        s = -(1.0f / kk) * expf(-(kl * kl) / kk) * expf(-(kk * kk) / ks4);
    }
    re[idx] *= s;
    im[idx] *= s;
}

// grad_k = i * (8 sin(w) - sin(2w))/6 * pot_k,  w = kvec[dir]
__global__ __launch_bounds__(256) void grad_kernel(const float* __restrict__ pre,
                                                   const float* __restrict__ pim,
                                                   float* __restrict__ ore,
                                                   float* __restrict__ oim, int dir) {
    int idx = blockIdx.x * blockDim.x + threadIdx.x;
    if (idx >= MESH_N3) return;
    int z = idx & 127, y = (idx >> 7) & 127, x = idx >> 14;
    int c = (dir == 0) ? x : ((dir == 1) ? y : z);
    float w = kfreq(c);
    float g = (8.0f * sinf(w) - sinf(2.0f * w)) * (1.0f / 6.0f);
    // i*g*(re + i*im) = -g*im + i*g*re
    ore[idx] = -g * pim[idx];
    oim[idx] = g * pre[idx];
}

// ---------------------------------------------------------------------------
// CIC read: 8-corner weighted gather; out[p*3+dir] = alpha * sum.
// ---------------------------------------------------------------------------
__global__ __launch_bounds__(256) void cic_read_kernel(const float* __restrict__ mesh,
                                                       const float* __restrict__ pos,
                                                       const float* __restrict__ alpha,
                                                       float* __restrict__ out, int dir) {
    int p = blockIdx.x * blockDim.x + threadIdx.x;
    if (p >= NPART) return;
    float px = pos[3 * p + 0];
    float py = pos[3 * p + 1];
    float pz = pos[3 * p + 2];
    float fx = floorf(px), fy = floorf(py), fz = floorf(pz);
    float acc = 0.0f;
#pragma unroll
    for (int c = 0; c < 8; ++c) {
        float nx = fx + (float)((c >> 2) & 1);
        float ny = fy + (float)((c >> 1) & 1);
        float nz = fz + (float)(c & 1);
        float w = (1.0f - fabsf(px - nx)) * (1.0f - fabsf(py - ny)) * (1.0f - fabsf(pz - nz));
        int ix = ((int)nx) & 127;
        int iy = ((int)ny) & 127;
        int iz = ((int)nz) & 127;
        acc += mesh[(ix << 14) + (iy << 7) + iz] * w;
    }
    out[3 * p + dir] = alpha[0] * acc;
}

// ---------------------------------------------------------------------------
// Orchestration
// ---------------------------------------------------------------------------
extern "C" void kernel_launch(void* const* d_in, const int* in_sizes, int n_in,
                              void* d_out, int out_size, void* d_ws, size_t ws_size,
                              hipStream_t stream) {
    (void)in_sizes; (void)n_in; (void)out_size; (void)ws_size;

    const float* pos   = (const float*)d_in[0];
    const float* alpha = (const float*)d_in[1];
    const float* kl    = (const float*)d_in[2];
    const float* ks    = (const float*)d_in[3];
    float* out = (float*)d_out;

    // Workspace layout (floats): 4 twiddle planes + 7 mesh planes = ~59 MB
    float* ws   = (float*)d_ws;
    float* Wfr  = ws;                    // 16384
    float* Wfi  = Wfr + MESH_N2;
    float* Wir  = Wfi + MESH_N2;
    float* Wii  = Wir + MESH_N2;
    float* mesh = Wii + MESH_N2;         // 128^3
    float* potR = mesh + MESH_N3;
    float* potI = potR + MESH_N3;
    float* Are  = potI + MESH_N3;
    float* Aim  = Are + MESH_N3;
    float* Bre  = Aim + MESH_N3;
    float* Bim  = Bre + MESH_N3;

    const dim3 blk(256);
    const dim3 gTw(MESH_N2 / 256);       // 64
    const dim3 gMesh(MESH_N3 / 256);     // 8192
    const dim3 gPart(NPART / 256);       // 16384
    const dim3 gDft(512);                // 8 col-tiles x 64 row-groups (16 tiles each)

    // 0) twiddles + zero mesh (deterministic per call; atomics need fresh zeros)
    twiddle_kernel<<<gTw, blk, 0, stream>>>(Wfr, Wfi, Wir, Wii);
    zero_kernel<<<gMesh, blk, 0, stream>>>(mesh, MESH_N3);

    // 1) CIC paint
    cic_paint_kernel<<<gPart, blk, 0, stream>>>(pos, mesh);

    // 2) forward 3-D DFT: mesh(real) -> A -> B -> pot  (delta_k)
    //    axis z: stride 1, rows over (x,y); axis y: stride 128, rows (x,z);
    //    axis x: stride 16384, rows (y,z).
    dft_pass<1, MESH_N2, MESH_N, false, true>
        <<<gDft, blk, 0, stream>>>(mesh, nullptr, Are, Aim, Wfr, Wfi);
    dft_pass<MESH_N, MESH_N2, 1, true, true>
        <<<gDft, blk, 0, stream>>>(Are, Aim, Bre, Bim, Wfr, Wfi);
    dft_pass<MESH_N2, MESH_N, 1, true, true>
        <<<gDft, blk, 0, stream>>>(Bre, Bim, potR, potI, Wfr, Wfi);

    // 3) pot_k = delta_k * invlaplace * pgd   (in place)
    pot_kernel<<<gMesh, blk, 0, stream>>>(potR, potI, kl, ks);

    // 4) per direction: gradient multiply, inverse 3-D DFT, CIC read
    for (int dir = 0; dir < 3; ++dir) {
        grad_kernel<<<gMesh, blk, 0, stream>>>(potR, potI, Are, Aim, dir);
        dft_pass<1, MESH_N2, MESH_N, true, true>
            <<<gDft, blk, 0, stream>>>(Are, Aim, Bre, Bim, Wir, Wii);
        dft_pass<MESH_N, MESH_N2, 1, true, true>
            <<<gDft, blk, 0, stream>>>(Bre, Bim, Are, Aim, Wir, Wii);
        dft_pass<MESH_N2, MESH_N, 1, true, false>
            <<<gDft, blk, 0, stream>>>(Are, Aim, Bre, Bim, Wir, Wii);
        cic_read_kernel<<<gPart, blk, 0, stream>>>(Bre, pos, alpha, out, dir);
    }
}